// EuclideanCodebook_31250182046394
// MI455X (gfx1250) — compile-verified
//
#include <hip/hip_runtime.h>
#include <math.h>

typedef __attribute__((ext_vector_type(2))) float v2f;
typedef __attribute__((ext_vector_type(4))) float v4f;
typedef __attribute__((ext_vector_type(8))) float v8f;

#define DIM 64
#define KCODES 2048
#define NROWS 131072
#define CHUNK 256            // codes staged in LDS per outer iteration
#define LDS_STRIDE 68        // floats per staged code row: 272B -> 16B aligned, bank stride 4 (conflict-free)
#define ROWS_PER_BLOCK 128   // 8 wave32s x 16 rows

// ---------------------------------------------------------------------------
// Phase 0: e_sq[k] = sum(embed[k,:]^2), zero histogram (ws is poisoned 0xAA)
// ---------------------------------------------------------------------------
__launch_bounds__(256)
__global__ void ec_prep_kernel(const float* __restrict__ embed,
                               float* __restrict__ e_sq,
                               unsigned* __restrict__ counts) {
    int k = blockIdx.x * blockDim.x + threadIdx.x;
    if (k < KCODES) {
        const float* row = embed + (size_t)k * DIM;
        float s = 0.f;
#pragma unroll
        for (int j = 0; j < DIM; j += 4) {
            v4f v = *(const v4f*)(row + j);
            s += v.x * v.x + v.y * v.y + v.z * v.z + v.w * v.w;
        }
        e_sq[k] = s;
        counts[k] = 0u;
    }
}

// ---------------------------------------------------------------------------
// Phase 1: nearest-code argmax via V_WMMA_F32_16X16X4_F32
//   score(n|row) = 2 * <x_row, e_n> - ||e_n||^2   (||x||^2 constant per row)
// A fragment (16x4 f32): lanes 0-15 hold K=+0,+1 ; lanes 16-31 hold K=+2,+3
// C/D (16x16 f32): lane = {half*16 + n}, vgpr v -> M = half*8 + v
// ---------------------------------------------------------------------------
__launch_bounds__(256)
__global__ void ec_argmax_kernel(const float* __restrict__ x,
                                 const float* __restrict__ embed,
                                 const float* __restrict__ e_sq,
                                 int* __restrict__ ind_out) {
    __shared__ float smem[CHUNK * LDS_STRIDE];

    const int tid  = threadIdx.x;
    const int wave = tid >> 5;
    const int lane = tid & 31;
    const int half = lane >> 4;   // 0: lanes 0-15, 1: lanes 16-31
    const int l16  = lane & 15;
    const int kb   = half * 2;    // per-lane k sub-offset within each 4-wide step

    const int rowbase = blockIdx.x * ROWS_PER_BLOCK + wave * 16;

    // Load the wave's 16x64 A panel once into 16 fragments (reused for all 2048 codes)
    v2f a[16];
    {
        const float* xr = x + (size_t)(rowbase + l16) * DIM + kb;
#pragma unroll
        for (int s = 0; s < 16; ++s) {
            a[s] = *(const v2f*)(xr + 4 * s);
        }
    }

    float best[8];
    int   bidx[8];
#pragma unroll
    for (int v = 0; v < 8; ++v) { best[v] = -3.402823466e38f; bidx[v] = 0; }

    for (int c = 0; c < KCODES; c += CHUNK) {
        __syncthreads();
        // Cooperative stage: thread t copies code row (c+t) into padded LDS
        {
            const float* er = embed + (size_t)(c + tid) * DIM;
            float* dst = smem + tid * LDS_STRIDE;
#pragma unroll
            for (int j = 0; j < DIM; j += 4) {
                *(v4f*)(dst + j) = *(const v4f*)(er + j);
            }
        }
        __syncthreads();

#pragma unroll 1
        for (int t = 0; t < CHUNK / 16; ++t) {
            const int nb = c + t * 16;
            v8f acc = {0.f, 0.f, 0.f, 0.f, 0.f, 0.f, 0.f, 0.f};
            const float* bsrc = smem + (t * 16 + l16) * LDS_STRIDE + kb;
#pragma unroll
            for (int s = 0; s < 16; ++s) {
                v2f b = *(const v2f*)(bsrc + 4 * s);
                acc = __builtin_amdgcn_wmma_f32_16x16x4_f32(
                    /*neg_a=*/false, a[s], /*neg_b=*/false, b,
                    /*c_mod=*/(short)0, acc, /*reuse_a=*/false, /*reuse_b=*/false);
            }
            const float es = e_sq[nb + l16];
            const int   ci = nb + l16;
#pragma unroll
            for (int v = 0; v < 8; ++v) {
                float val = 2.0f * acc[v] - es;
                if (val > best[v]) { best[v] = val; bidx[v] = ci; }  // strict > keeps lowest index
            }
        }
    }

    // Reduce across each 16-lane half (column dimension); lower index wins ties
#pragma unroll
    for (int v = 0; v < 8; ++v) {
        float bv = best[v];
        int   bi = bidx[v];
#pragma unroll
        for (int m = 1; m <= 8; m <<= 1) {
            float ov = __shfl_xor(bv, m, 32);
            int   oi = __shfl_xor(bi, m, 32);
            if (ov > bv || (ov == bv && oi < bi)) { bv = ov; bi = oi; }
        }
        if (l16 == 0) ind_out[rowbase + half * 8 + v] = bi;
    }
}

// ---------------------------------------------------------------------------
// Phase 2: quantize = embed[ind], embed_ind (as f32), histogram counts
// 64 lanes per row, 4 rows per 256-thread block
// ---------------------------------------------------------------------------
__launch_bounds__(256)
__global__ void ec_gather_kernel(const float* __restrict__ embed,
                                 const int* __restrict__ ind,
                                 float* __restrict__ quant_out,
                                 float* __restrict__ ind_out,
                                 unsigned* __restrict__ counts) {
    const int tid = threadIdx.x;
    const int row = blockIdx.x * 4 + (tid >> 6);
    const int j   = tid & 63;
    const int k   = ind[row];
    quant_out[(size_t)row * DIM + j] = embed[(size_t)k * DIM + j];
    if (j == 0) {
        ind_out[row] = (float)k;
        atomicAdd(&counts[k], 1u);
    }
}

// ---------------------------------------------------------------------------
// Phase 3: perplexity = exp(-sum(avg * log(avg + 1e-10)))
// ---------------------------------------------------------------------------
__launch_bounds__(256)
__global__ void ec_perplexity_kernel(const unsigned* __restrict__ counts,
                                     float* __restrict__ out) {
    __shared__ float red[256];
    float s = 0.f;
    for (int k = threadIdx.x; k < KCODES; k += 256) {
        float avg = (float)counts[k] * (1.0f / (float)NROWS);
        s += avg * logf(avg + 1e-10f);
    }
    red[threadIdx.x] = s;
    __syncthreads();
    for (int off = 128; off > 0; off >>= 1) {
        if (threadIdx.x < off) red[threadIdx.x] += red[threadIdx.x + off];
        __syncthreads();
    }
    if (threadIdx.x == 0) out[0] = expf(-red[0]);
}

// ---------------------------------------------------------------------------
extern "C" void kernel_launch(void* const* d_in, const int* in_sizes, int n_in,
                              void* d_out, int out_size, void* d_ws, size_t ws_size,
                              hipStream_t stream) {
    (void)in_sizes; (void)n_in; (void)out_size; (void)ws_size;

    const float* x     = (const float*)d_in[0];   // [64,2048,64] f32 -> N=131072 rows
    const float* embed = (const float*)d_in[1];   // [2048,64] f32

    float* quant     = (float*)d_out;                       // NROWS*DIM
    float* ind_out_f = quant + (size_t)NROWS * DIM;         // NROWS
    float* perp      = ind_out_f + NROWS;                   // 1

    int*      ind_ws = (int*)d_ws;                                      // NROWS ints
    float*    e_sq   = (float*)((char*)d_ws + (size_t)NROWS * sizeof(int)); // KCODES f32
    unsigned* counts = (unsigned*)(e_sq + KCODES);                      // KCODES u32

    ec_prep_kernel<<<(KCODES + 255) / 256, 256, 0, stream>>>(embed, e_sq, counts);
    ec_argmax_kernel<<<NROWS / ROWS_PER_BLOCK, 256, 0, stream>>>(x, embed, e_sq, ind_ws);
    ec_gather_kernel<<<NROWS / 4, 256, 0, stream>>>(embed, ind_ws, quant, ind_out_f, counts);
    ec_perplexity_kernel<<<1, 256, 0, stream>>>(counts, perp);
}